// SAGE_81484119540292
// MI455X (gfx1250) — compile-verified
//
#include <hip/hip_runtime.h>
#include <hip/hip_bf16.h>
#include <stddef.h>

// ---------------------------------------------------------------------------
// GraphSAGE 3-layer inference for MI455X (gfx1250, wave32).
//  - sparse mean-aggregation via global_atomic_add_f32
//  - dense GEMMs via V_WMMA_F32_16X16X4_F32 (fp32-exact vs reference)
//  - weights pre-transposed so B fragments are single b64 loads
//  - 64 rows (4 M-tiles) per block: each B fragment feeds 4 WMMAs
// ---------------------------------------------------------------------------

typedef __attribute__((ext_vector_type(2))) float v2f;
typedef __attribute__((ext_vector_type(8))) float v8f;

#define DFEAT 128   // K dimension of every GEMM in this model

// ---------------------------------------------------------------------------
__global__ void deg_kernel(const int* __restrict__ dst, float* __restrict__ deg,
                           int E) {
    int e = blockIdx.x * blockDim.x + threadIdx.x;
    if (e < E) atomicAdd(&deg[dst[e]], 1.0f);
}

// deg -> 1/max(deg,1), in place
__global__ void invdeg_kernel(float* __restrict__ deg, int N) {
    int v = blockIdx.x * blockDim.x + threadIdx.x;
    if (v < N) {
        float d = deg[v];
        deg[v] = 1.0f / (d > 1.0f ? d : 1.0f);
    }
}

// wt[c][k] = w[k][c]   (w is [K][D] row-major)
__global__ void transpose_kernel(const float* __restrict__ w,
                                 float* __restrict__ wt, int K, int D) {
    int i = blockIdx.x * blockDim.x + threadIdx.x;
    if (i < K * D) {
        int k = i / D, c = i % D;
        wt[(size_t)c * K + k] = w[i];
    }
}

// ---------------------------------------------------------------------------
// agg[dst[e]][:] += x[src[e]][:]   (one wave32 per edge, float4 per lane)
// ---------------------------------------------------------------------------
__global__ void scatter_kernel(const float* __restrict__ x,
                               const int* __restrict__ src,
                               const int* __restrict__ dst,
                               float* __restrict__ agg, int E) {
    int e = blockIdx.x * (blockDim.x >> 5) + (threadIdx.x >> 5);
    if (e >= E) return;
    int lane = threadIdx.x & 31;
    int s = src[e];
    int d = dst[e];
    const float4 v = ((const float4*)(x + (size_t)s * DFEAT))[lane];
    float* a = agg + (size_t)d * DFEAT + lane * 4;
    atomicAdd(a + 0, v.x);
    atomicAdd(a + 1, v.y);
    atomicAdd(a + 2, v.z);
    atomicAdd(a + 3, v.w);
}

// ---------------------------------------------------------------------------
// out[64-row tile] = act( x @ Wself + b  [+ (agg*invdeg) @ Wneigh] )
//
// Block: DOUT/16 waves (wave w owns output cols [16w,16w+16)), 64 rows =
// 4 M-tiles. One shared 32KB buffer staged twice (self pass, neighbor pass).
// Weights come in TRANSPOSED: wT[col][k], so a B fragment is one b64 load.
//
// WMMA fp32 fragment layouts (CDNA5 ISA 7.12.2):
//   A 16x4 : lane l -> row l&15, K = k0 + 2*(l>>4) + {0,1}   (2 VGPRs)
//   B 4x16 : lane l -> col l&15, K = k0 + 2*(l>>4) + {0,1}   (2 VGPRs)
//   C 16x16: lane l, vgpr j -> row j + 8*(l>>4), col l&15    (8 VGPRs)
// ---------------------------------------------------------------------------
template <int DOUT, bool HAS_NEIGH, bool RELU>
__global__ void sage_wmma_kernel(const float* __restrict__ x,
                                 const float* __restrict__ agg,
                                 const float* __restrict__ invdeg,
                                 const float* __restrict__ wselfT,   // [DOUT][128]
                                 const float* __restrict__ bias,     // [DOUT]
                                 const float* __restrict__ wneighT,  // [DOUT][128]
                                 float* __restrict__ out,            // [N][DOUT]
                                 int N) {
    constexpr int MT = 4;            // M-tiles per block
    constexpr int ROWS = MT * 16;    // 64 rows
    __shared__ __align__(16) float sm[ROWS * DFEAT];   // 32 KB

    const int row0 = blockIdx.x * ROWS;
    const int tid  = threadIdx.x;
    constexpr int NTHR = DOUT * 2;   // DOUT/16 waves

    const int wave = tid >> 5;
    const int lane = tid & 31;
    const int m    = lane & 15;            // A-fragment row within tile
    const int koff = (lane >> 4) << 1;     // 0 or 2
    const int col  = wave * 16 + (lane & 15);

    // ---- stage 64x128 self tile ----
    for (int i = tid; i < ROWS * DFEAT; i += NTHR) {
        int r = i >> 7;
        int rr = row0 + r;
        if (rr >= N) rr = N - 1;
        sm[i] = x[(size_t)rr * DFEAT + (i & (DFEAT - 1))];
    }
    __syncthreads();

    v8f acc[MT] = {};

    // ---- self GEMM pass ----
    {
        const float* bcol = wselfT + (size_t)col * DFEAT + koff;
        for (int k0 = 0; k0 < DFEAT; k0 += 4) {
            v2f b = *(const v2f*)(bcol + k0);
#pragma unroll
            for (int t = 0; t < MT; ++t) {
                v2f a = *(const v2f*)(&sm[(t * 16 + m) * DFEAT + k0 + koff]);
                acc[t] = __builtin_amdgcn_wmma_f32_16x16x4_f32(
                    false, a, false, b, (short)0, acc[t], false, false);
            }
        }
    }

    // ---- neighbor GEMM pass (same LDS buffer, restaged) ----
    if constexpr (HAS_NEIGH) {
        __syncthreads();   // everyone done reading the self tile
        for (int i = tid; i < ROWS * DFEAT; i += NTHR) {
            int r = i >> 7;
            int rr = row0 + r;
            if (rr >= N) rr = N - 1;
            sm[i] = agg[(size_t)rr * DFEAT + (i & (DFEAT - 1))] * invdeg[rr];
        }
        __syncthreads();
        const float* bcol = wneighT + (size_t)col * DFEAT + koff;
        for (int k0 = 0; k0 < DFEAT; k0 += 4) {
            v2f b = *(const v2f*)(bcol + k0);
#pragma unroll
            for (int t = 0; t < MT; ++t) {
                v2f a = *(const v2f*)(&sm[(t * 16 + m) * DFEAT + k0 + koff]);
                acc[t] = __builtin_amdgcn_wmma_f32_16x16x4_f32(
                    false, a, false, b, (short)0, acc[t], false, false);
            }
        }
    }

    // ---- epilogue: bias, optional ReLU, store ----
    const float bv = bias[col];
    const int rh = (lane >> 4) << 3;   // 0 or 8
#pragma unroll
    for (int t = 0; t < MT; ++t) {
#pragma unroll
        for (int j = 0; j < 8; ++j) {
            int r = row0 + t * 16 + j + rh;
            if (r < N) {
                float v = acc[t][j] + bv;
                if (RELU) v = fmaxf(v, 0.0f);
                out[(size_t)r * DOUT + col] = v;
            }
        }
    }
}

// ---------------------------------------------------------------------------
extern "C" void kernel_launch(void* const* d_in, const int* in_sizes, int n_in,
                              void* d_out, int out_size, void* d_ws, size_t ws_size,
                              hipStream_t stream) {
    const float* x        = (const float*)d_in[0];
    const int*   src      = (const int*)d_in[1];
    const int*   dst      = (const int*)d_in[2];
    const float* w_self0  = (const float*)d_in[3];
    const float* b_self0  = (const float*)d_in[4];
    const float* w_neigh0 = (const float*)d_in[5];
    const float* fc_w     = (const float*)d_in[6];
    const float* fc_b     = (const float*)d_in[7];
    const float* fc2_w    = (const float*)d_in[8];
    const float* fc2_b    = (const float*)d_in[9];
    const float* w_self1  = (const float*)d_in[10];
    const float* b_self1  = (const float*)d_in[11];
    const float* w_neigh1 = (const float*)d_in[12];
    const float* w_self2  = (const float*)d_in[13];
    const float* b_self2  = (const float*)d_in[14];
    const float* w_neigh2 = (const float*)d_in[15];
    float* out = (float*)d_out;

    const int N = in_sizes[0] / DFEAT;       // 100000
    const int E = in_sizes[1];               // 600000
    const int DOUT2 = in_sizes[13] / DFEAT;  // 64 (unused except sanity)
    (void)DOUT2;

    // ---- workspace layout ----
    float* deg = (float*)d_ws;                     // N (becomes invdeg)
    float* agg = deg + N;                          // N*128
    float* h   = agg + (size_t)N * DFEAT;          // N*128
    float* wp  = h + (size_t)N * DFEAT;            // transposed weights
    float* wself0T  = wp;               wp += 128 * 128;
    float* wneigh0T = wp;               wp += 128 * 128;
    float* fcwT     = wp;               wp += 128 * 128;
    float* fc2wT    = wp;               wp += 128 * 128;
    float* wself1T  = wp;               wp += 128 * 128;
    float* wneigh1T = wp;               wp += 128 * 128;
    float* wself2T  = wp;               wp += 128 * 64;
    float* wneigh2T = wp;               wp += 128 * 64;

    const int ntile = (N + 63) / 64;
    const int scat_blocks = (E + 7) / 8;           // 8 edges (waves) per block
    const dim3 b256(256), b128(128);

    // ---- weight transposes + degree (independent of layers) ----
    transpose_kernel<<<(128 * 128 + 255) / 256, b256, 0, stream>>>(w_self0,  wself0T,  128, 128);
    transpose_kernel<<<(128 * 128 + 255) / 256, b256, 0, stream>>>(w_neigh0, wneigh0T, 128, 128);
    transpose_kernel<<<(128 * 128 + 255) / 256, b256, 0, stream>>>(fc_w,     fcwT,     128, 128);
    transpose_kernel<<<(128 * 128 + 255) / 256, b256, 0, stream>>>(fc2_w,    fc2wT,    128, 128);
    transpose_kernel<<<(128 * 128 + 255) / 256, b256, 0, stream>>>(w_self1,  wself1T,  128, 128);
    transpose_kernel<<<(128 * 128 + 255) / 256, b256, 0, stream>>>(w_neigh1, wneigh1T, 128, 128);
    transpose_kernel<<<(128 * 64 + 255) / 256,  b256, 0, stream>>>(w_self2,  wself2T,  128, 64);
    transpose_kernel<<<(128 * 64 + 255) / 256,  b256, 0, stream>>>(w_neigh2, wneigh2T, 128, 64);

    hipMemsetAsync(deg, 0, (size_t)N * sizeof(float), stream);
    deg_kernel<<<(E + 255) / 256, b256, 0, stream>>>(dst, deg, E);
    invdeg_kernel<<<(N + 255) / 256, b256, 0, stream>>>(deg, N);

    // ---- layer 0: SAGEConv(128->128)+ReLU, fc+ReLU, fc2+ReLU ----
    hipMemsetAsync(agg, 0, (size_t)N * DFEAT * sizeof(float), stream);
    scatter_kernel<<<scat_blocks, b256, 0, stream>>>(x, src, dst, agg, E);
    sage_wmma_kernel<128, true, true><<<ntile, b256, 0, stream>>>(
        x, agg, deg, wself0T, b_self0, wneigh0T, h, N);
    sage_wmma_kernel<128, false, true><<<ntile, b256, 0, stream>>>(
        h, nullptr, nullptr, fcwT, fc_b, nullptr, h, N);
    sage_wmma_kernel<128, false, true><<<ntile, b256, 0, stream>>>(
        h, nullptr, nullptr, fc2wT, fc2_b, nullptr, h, N);

    // ---- layer 1: SAGEConv(128->128)+ReLU ----
    hipMemsetAsync(agg, 0, (size_t)N * DFEAT * sizeof(float), stream);
    scatter_kernel<<<scat_blocks, b256, 0, stream>>>(h, src, dst, agg, E);
    sage_wmma_kernel<128, true, true><<<ntile, b256, 0, stream>>>(
        h, agg, deg, wself1T, b_self1, wneigh1T, h, N);

    // ---- layer 2: SAGEConv(128->64), no activation ----
    hipMemsetAsync(agg, 0, (size_t)N * DFEAT * sizeof(float), stream);
    scatter_kernel<<<scat_blocks, b256, 0, stream>>>(h, src, dst, agg, E);
    sage_wmma_kernel<64, true, false><<<ntile, b128, 0, stream>>>(
        h, agg, deg, wself2T, b_self2, wneigh2T, out, N);
}